// MetaDynamics_53377853554903
// MI455X (gfx1250) — compile-verified
//
#include <hip/hip_runtime.h>
#include <hip/hip_bf16.h>
#include <math.h>

typedef __attribute__((ext_vector_type(2))) float v2f;
typedef __attribute__((ext_vector_type(8))) float v8f;

#define NHILLS 16384
#define NPTS   4096
#define NCV    8
#define KPAD   20          // 17 useful K slots padded to multiple of 4
#define KPAIRS (KPAD / 2)  // float2 pairs
#define HSPLIT 8           // hill-dimension grid split
#define WAVES  8           // waves per block
// tiles: 1024 hill-tiles total; per block 1024/HSPLIT = 128; per wave 128/WAVES = 16
#define TILES_PER_WAVE ((NHILLS / 16) / HSPLIT / WAVES)

// -0.5 * log2(e): scale so WMMA output is the base-2 exponent directly
#define SCL (-0.72134752044448169f)

__device__ __forceinline__ float fast_exp2(float x) {
#if defined(__HIP_DEVICE_COMPILE__) && __has_builtin(__builtin_amdgcn_exp2f)
  return __builtin_amdgcn_exp2f(x);   // v_exp_f32
#else
  return exp2f(x);
#endif
}

// ---- Prep: per-hill augmented A rows, stored transposed as k-pairs [kp][h] ----
__global__ void md_prep_hills(const float* __restrict__ cen,
                              const float* __restrict__ wdt,
                              const float* __restrict__ hgt,
                              v2f* __restrict__ hillP) {
  int h = blockIdx.x * blockDim.x + threadIdx.x;
  if (h >= NHILLS) return;
  float k[KPAD];
  float base = 0.f;
#pragma unroll
  for (int d = 0; d < NCV; ++d) {
    float w = wdt[h * NCV + d];
    float c = cen[h * NCV + d];
    float inv = 1.0f / (w * w);
    k[d]       = SCL * inv;                 // multiplies col^2
    k[NCV + d] = SCL * (-2.0f * c * inv);   // multiplies col
    base += c * c * inv;
  }
  // fold per-hill constant AND height into the K=16 slot (times 1.0 on B side)
  k[16] = SCL * base + __log2f(hgt[h]);     // hgt==0 -> -inf -> exp2 -> 0 (correct)
  k[17] = 0.f; k[18] = 0.f; k[19] = 0.f;
#pragma unroll
  for (int j = 0; j < KPAIRS; ++j) {
    v2f p; p.x = k[2 * j]; p.y = k[2 * j + 1];
    hillP[j * NHILLS + h] = p;
  }
}

// ---- Prep: per-point augmented B columns, stored transposed as k-pairs [kp][p] ----
__global__ void md_prep_points(const float* __restrict__ col,
                               v2f* __restrict__ ptP) {
  int p = blockIdx.x * blockDim.x + threadIdx.x;
  if (p >= NPTS) return;
  float k[KPAD];
#pragma unroll
  for (int d = 0; d < NCV; ++d) {
    float c = col[p * NCV + d];
    k[d]       = c * c;
    k[NCV + d] = c;
  }
  k[16] = 1.0f;
  k[17] = 0.f; k[18] = 0.f; k[19] = 0.f;
#pragma unroll
  for (int j = 0; j < KPAIRS; ++j) {
    v2f q; q.x = k[2 * j]; q.y = k[2 * j + 1];
    ptP[j * NPTS + p] = q;
  }
}

__global__ void md_zero_out(float* __restrict__ out) {
  int p = blockIdx.x * blockDim.x + threadIdx.x;
  if (p < NPTS) out[p] = 0.0f;
}

// ---- Main: tiled 16x16 WMMA (fp32, K=4) + exp2 reduction ----
// grid = (NPTS/16, HSPLIT), block = 32*WAVES
__global__ __launch_bounds__(32 * WAVES)
void md_main(const v2f* __restrict__ hillP,
             const v2f* __restrict__ ptP,
             float* __restrict__ out) {
  __shared__ float red[WAVES][16];

  const int lane = threadIdx.x & 31;
  const int wave = threadIdx.x >> 5;
  const int m  = lane & 15;       // row/col within tile for this lane
  const int kl = lane >> 4;       // lane-half -> K pair offset (K +0 vs +2)
  const int p0 = blockIdx.x * 16;

  // B fragments for this block's 16 points: fragment j covers K = 4j..4j+3
  // VGPR layout assumption (mirrors documented fp32 A 16x4 layout):
  //   .x holds K = 4j + 2*kl, .y holds K = 4j + 2*kl + 1
  v2f b0 = ptP[(0 + kl) * NPTS + p0 + m];
  v2f b1 = ptP[(2 + kl) * NPTS + p0 + m];
  v2f b2 = ptP[(4 + kl) * NPTS + p0 + m];
  v2f b3 = ptP[(6 + kl) * NPTS + p0 + m];
  v2f b4 = ptP[(8 + kl) * NPTS + p0 + m];

  const int tbase = blockIdx.y * (TILES_PER_WAVE * WAVES) + wave * TILES_PER_WAVE;

  float acc = 0.0f;
#pragma unroll 2
  for (int t = 0; t < TILES_PER_WAVE; ++t) {
    const int h0 = (tbase + t) << 4;
    v2f a0 = hillP[(0 + kl) * NHILLS + h0 + m];
    v2f a1 = hillP[(2 + kl) * NHILLS + h0 + m];
    v2f a2 = hillP[(4 + kl) * NHILLS + h0 + m];
    v2f a3 = hillP[(6 + kl) * NHILLS + h0 + m];
    v2f a4 = hillP[(8 + kl) * NHILLS + h0 + m];

    v8f c = {};
    c = __builtin_amdgcn_wmma_f32_16x16x4_f32(false, a0, false, b0, (short)0, c, false, false);
    c = __builtin_amdgcn_wmma_f32_16x16x4_f32(false, a1, false, b1, (short)0, c, false, false);
    c = __builtin_amdgcn_wmma_f32_16x16x4_f32(false, a2, false, b2, (short)0, c, false, false);
    c = __builtin_amdgcn_wmma_f32_16x16x4_f32(false, a3, false, b3, (short)0, c, false, false);
    c = __builtin_amdgcn_wmma_f32_16x16x4_f32(false, a4, false, b4, (short)0, c, false, false);

    // c[i] = log2 of the i-th hill-row's contribution for point p0+m.
    // All 8 elements in this lane belong to the same point (N = lane%16).
#pragma unroll
    for (int i = 0; i < 8; ++i) acc += fast_exp2(c[i]);
  }

  // lanes L and L+16 hold the same point's partial sums (different M halves)
  acc += __shfl_xor(acc, 16, 32);
  if (lane < 16) red[wave][m] = acc;
  __syncthreads();

  if (threadIdx.x < 16) {
    float s = 0.0f;
#pragma unroll
    for (int w = 0; w < WAVES; ++w) s += red[w][threadIdx.x];
    atomicAdd(&out[p0 + threadIdx.x], s);  // global_atomic_add_f32
  }
}

extern "C" void kernel_launch(void* const* d_in, const int* in_sizes, int n_in,
                              void* d_out, int out_size, void* d_ws, size_t ws_size,
                              hipStream_t stream) {
  const float* col = (const float*)d_in[0];  // [NPTS,  NCV]
  const float* cen = (const float*)d_in[1];  // [NHILLS,NCV]
  const float* wdt = (const float*)d_in[2];  // [NHILLS,NCV]
  const float* hgt = (const float*)d_in[3];  // [NHILLS]
  float* out = (float*)d_out;

  v2f* hillP = (v2f*)d_ws;                                        // KPAIRS*NHILLS float2
  v2f* ptP   = (v2f*)((char*)d_ws + (size_t)KPAIRS * NHILLS * 8); // KPAIRS*NPTS float2

  md_prep_hills<<<NHILLS / 256, 256, 0, stream>>>(cen, wdt, hgt, hillP);
  md_prep_points<<<NPTS / 256, 256, 0, stream>>>(col, ptP);
  md_zero_out<<<NPTS / 256, 256, 0, stream>>>(out);

  dim3 grid(NPTS / 16, HSPLIT);
  md_main<<<grid, 32 * WAVES, 0, stream>>>(hillP, ptP, out);
}